// Attention_37649683317367
// MI455X (gfx1250) — compile-verified
//
#include <hip/hip_runtime.h>
#include <stdint.h>

#define NUM_HEADS 32
#define HEAD_SIZE 128
#define HIDDEN (NUM_HEADS * HEAD_SIZE)
#define MAX_SEQ 4096
#define NEG_BIG (-1e30f)
#define ATTN_SCALE 0.088388347648318447f   /* 1/sqrt(128) */

#define CHUNK   16
#define WAVES   4
#define BLOCKT  (WAVES * 32)
#define KSTRIDE 132   /* floats; 16-row K/V chunk row stride, conflict-free b64 reads */
#define PSTRIDE 20    /* floats; P transpose scratch row stride, conflict-free b64 reads */

typedef float v2f __attribute__((ext_vector_type(2)));
typedef float v8f __attribute__((ext_vector_type(8)));

__global__ __launch_bounds__(BLOCKT) void attn_fa_f32wmma_kernel(
    const float* __restrict__ query,
    const float* __restrict__ key,
    const float* __restrict__ value,
    const float* __restrict__ kv_cache,
    const int*   __restrict__ offset_ptr,
    float*       __restrict__ out,
    int seq_len)
{
  // double-buffered K/V chunk staging + per-wave P transpose scratch
  __shared__ __align__(16) float Ksh[2][CHUNK * KSTRIDE];
  __shared__ __align__(16) float Vsh[2][CHUNK * KSTRIDE];
  __shared__ __align__(16) float Psh[WAVES * 16 * PSTRIDE];

  const int head = blockIdx.x;        // 0..31
  const int qblk = blockIdx.y;        // 64 query rows per block
  const int tid  = threadIdx.x;
  const int wave = tid >> 5;
  const int lane = tid & 31;
  const int half = lane >> 4;         // 0 or 1 (wave32 half)
  const int l16  = lane & 15;

  const int offset  = offset_ptr[0];
  const int cur_pos = offset + seq_len;
  const int qbase   = qblk * 64 + wave * 16;   // this wave's query tile (global row)

  // ---- load Q tile as WMMA A-fragments, pre-scaled by 1/sqrt(d) ----
  // A layout (16x4 f32): lane -> M = lane%16; vgpr j -> K = 2*(lane/16) + j
  v2f qfrag[32];
  {
    const int  qrow   = qbase + l16;
    const bool qvalid = (qrow < seq_len);
    const int  qclamp = qvalid ? qrow : (seq_len - 1);
    const float* qp = query + (size_t)qclamp * HIDDEN + head * HEAD_SIZE + 2 * half;
#pragma unroll
    for (int kk = 0; kk < 32; ++kk) {
      float2 q2 = *(const float2*)(qp + 4 * kk);
      qfrag[kk][0] = qvalid ? q2.x * ATTN_SCALE : 0.0f;
      qfrag[kk][1] = qvalid ? q2.y * ATTN_SCALE : 0.0f;
    }
  }

  // ---- accumulators: O (16x128) as 8 C/D tiles; online-softmax state ----
  const v8f vzero = {0.f, 0.f, 0.f, 0.f, 0.f, 0.f, 0.f, 0.f};
  v8f acc[8];
#pragma unroll
  for (int i = 0; i < 8; ++i) acc[i] = vzero;
  float mrun[8], lsum[8];
#pragma unroll
  for (int r = 0; r < 8; ++r) { mrun[r] = NEG_BIG; lsum[r] = 0.0f; }

  // context needed by this block: tokens [0, offset + qblk*64 + 64)
  int ctx_len = offset + qblk * 64 + 64;
  if (ctx_len > cur_pos) ctx_len = cur_pos;
  const int nchunks = (ctx_len + CHUNK - 1) / CHUNK;

  const int t_loc = tid >> 3;         // 0..15: chunk row this thread stages
  const int dseg  = (tid & 7) * 16;   // 16-float dim segment (coalesced global reads)
  const int qpos0 = offset + qbase + 8 * half;  // abs query pos of this lane's r=0 row

  // LDS byte offsets of this thread's async-staging destinations, per buffer
  const uint32_t koff[2] = { (uint32_t)(uintptr_t)&Ksh[0][t_loc * KSTRIDE + dseg],
                             (uint32_t)(uintptr_t)&Ksh[1][t_loc * KSTRIDE + dseg] };
  const uint32_t voff[2] = { (uint32_t)(uintptr_t)&Vsh[0][t_loc * KSTRIDE + dseg],
                             (uint32_t)(uintptr_t)&Vsh[1][t_loc * KSTRIDE + dseg] };

  // issue the 8 async LDS fills (4x K, 4x V) for one chunk into one buffer
  auto stage = [&](int c, int buf) {
    int t = c * CHUNK + t_loc;
    if (t >= cur_pos) t = cur_pos - 1;   // clamp; such tokens are causally masked anyway
    const float *ksrc, *vsrc;
    if (t < offset) {
      ksrc = kv_cache + (size_t)t * HIDDEN + head * HEAD_SIZE;
      vsrc = kv_cache + (size_t)(MAX_SEQ + t) * HIDDEN + head * HEAD_SIZE;
    } else {
      const int tr = t - offset;
      ksrc = key   + (size_t)tr * HIDDEN + head * HEAD_SIZE;
      vsrc = value + (size_t)tr * HIDDEN + head * HEAD_SIZE;
    }
    const uint64_t kga = (uint64_t)(uintptr_t)(ksrc + dseg);
    const uint64_t vga = (uint64_t)(uintptr_t)(vsrc + dseg);
    asm volatile(
      "global_load_async_to_lds_b128 %0, %2, off\n\t"
      "global_load_async_to_lds_b128 %0, %2, off offset:16\n\t"
      "global_load_async_to_lds_b128 %0, %2, off offset:32\n\t"
      "global_load_async_to_lds_b128 %0, %2, off offset:48\n\t"
      "global_load_async_to_lds_b128 %1, %3, off\n\t"
      "global_load_async_to_lds_b128 %1, %3, off offset:16\n\t"
      "global_load_async_to_lds_b128 %1, %3, off offset:32\n\t"
      "global_load_async_to_lds_b128 %1, %3, off offset:48"
      :
      : "v"(koff[buf]), "v"(voff[buf]), "v"(kga), "v"(vga)
      : "memory");
  };

  // prologue: fill buffer 0 with chunk 0
  stage(0, 0);

  for (int c = 0; c < nchunks; ++c) {
    const int   cbase = c * CHUNK;
    const int   buf   = c & 1;
    const float* Kb = Ksh[buf];
    const float* Vb = Vsh[buf];

    // prefetch next chunk into the other buffer, then wait only for the OLD group
    // (async loads complete in order: <=8 outstanding => this chunk's 8 are done)
    if (c + 1 < nchunks) {
      stage(c + 1, buf ^ 1);
      asm volatile("s_wait_asynccnt 0x8" ::: "memory");
    } else {
      asm volatile("s_wait_asynccnt 0x0" ::: "memory");
    }
    __syncthreads();   // all waves' fills for `buf` complete before anyone reads it

    // ---- S(16x16) = Q * K^T : 32 f32 WMMAs over d=128, two independent chains ----
    // B layout (4x16): lane -> N = lane%16 (token); vgpr j -> K = 2*(lane/16)+j (dim)
    v8f s0 = vzero, s1 = vzero;
#pragma unroll
    for (int g = 0; g < 4; ++g) {
      v2f kf[8];
#pragma unroll
      for (int j = 0; j < 8; ++j) {
        const float2 k2 = *(const float2*)&Kb[l16 * KSTRIDE + 4 * (8 * g + j) + 2 * half];
        kf[j][0] = k2.x; kf[j][1] = k2.y;
      }
#pragma unroll
      for (int j = 0; j < 8; j += 2) {
        s0 = __builtin_amdgcn_wmma_f32_16x16x4_f32(false, qfrag[8 * g + j],     false, kf[j],
                                                   (short)0, s0, false, false);
        s1 = __builtin_amdgcn_wmma_f32_16x16x4_f32(false, qfrag[8 * g + j + 1], false, kf[j + 1],
                                                   (short)0, s1, false, false);
      }
    }
    const v8f s = s0 + s1;

    // ---- causal mask + online softmax (row m = 8*half + r, col token = cbase+l16) ----
    const int t_tok = cbase + l16;
    float pv[8], corr[8];
#pragma unroll
    for (int r = 0; r < 8; ++r) {
      float sv = (t_tok <= qpos0 + r) ? s[r] : NEG_BIG;
      float rmax = sv;
#pragma unroll
      for (int m = 1; m < 16; m <<= 1) rmax = fmaxf(rmax, __shfl_xor(rmax, m, 32));
      const float mnew = fmaxf(mrun[r], rmax);
      const float cr   = __expf(mrun[r] - mnew);
      const float p    = __expf(sv - mnew);
      float rs = p;
#pragma unroll
      for (int m = 1; m < 16; m <<= 1) rs += __shfl_xor(rs, m, 32);
      lsum[r] = lsum[r] * cr + rs;
      mrun[r] = mnew;
      pv[r] = p; corr[r] = cr;
    }
#pragma unroll
    for (int dt = 0; dt < 8; ++dt) {
#pragma unroll
      for (int r = 0; r < 8; ++r) acc[dt][r] *= corr[r];
    }

    // ---- transpose P (D layout -> A layout) via per-wave LDS scratch ----
    float* pw = Psh + wave * 16 * PSTRIDE;
#pragma unroll
    for (int r = 0; r < 8; ++r) pw[(8 * half + r) * PSTRIDE + l16] = pv[r];
    __builtin_amdgcn_wave_barrier();  // same-wave DS ops are in-order; just pin scheduling

    // ---- O(16x128) += P(16x16) * V(16x128): 4 K-steps x 8 interleaved acc chains ----
#pragma unroll
    for (int kk = 0; kk < 4; ++kk) {
      const float2 p2 = *(const float2*)&pw[l16 * PSTRIDE + 4 * kk + 2 * half];
      v2f a; a[0] = p2.x; a[1] = p2.y;
      const int t0 = 4 * kk + 2 * half;
#pragma unroll
      for (int dt = 0; dt < 8; ++dt) {
        v2f b;
        b[0] = Vb[(t0 + 0) * KSTRIDE + dt * 16 + l16];
        b[1] = Vb[(t0 + 1) * KSTRIDE + dt * 16 + l16];
        acc[dt] = __builtin_amdgcn_wmma_f32_16x16x4_f32(false, a, false, b,
                                                        (short)0, acc[dt], false, false);
      }
    }
    __syncthreads();  // all waves done reading `buf` => safe to refill next iteration
  }

  // ---- finalize: divide by softmax denominator and write out ----
#pragma unroll
  for (int r = 0; r < 8; ++r) {
    const int qrow = qbase + 8 * half + r;
    if (qrow < seq_len) {
      const float inv = 1.0f / lsum[r];
      float* orow = out + (size_t)qrow * HIDDEN + head * HEAD_SIZE;
#pragma unroll
      for (int dt = 0; dt < 8; ++dt)
        orow[dt * 16 + l16] = acc[dt][r] * inv;
    }
  }
}

extern "C" void kernel_launch(void* const* d_in, const int* in_sizes, int n_in,
                              void* d_out, int out_size, void* d_ws, size_t ws_size,
                              hipStream_t stream) {
  const float* query = (const float*)d_in[0];
  const float* key   = (const float*)d_in[1];
  const float* value = (const float*)d_in[2];
  const float* kvc   = (const float*)d_in[3];
  const int*   off   = (const int*)d_in[4];
  const int seq_len  = in_sizes[0] / HIDDEN;

  dim3 grid(NUM_HEADS, (seq_len + 63) / 64);
  attn_fa_f32wmma_kernel<<<grid, BLOCKT, 0, stream>>>(query, key, value, kvc, off,
                                                      (float*)d_out, seq_len);
}